// FragmentGraphEncoder_25314537242759
// MI455X (gfx1250) — compile-verified
//
#include <hip/hip_runtime.h>
#include <hip/hip_bf16.h>

#define N_NODES 50000
#define E_EDGES 800000
#define IN_DIM  16
#define H_DIM   512
#define L_LAYERS 3
#define G_GRAPHS 256
#define LN_EPS  1e-5f

typedef __attribute__((ext_vector_type(2))) float v2f;
typedef __attribute__((ext_vector_type(8))) float v8f;

__device__ __forceinline__ v8f wmma_f32_k4(v2f a, v2f b, v8f c) {
  // D = A(16x4 f32) * B(4x16 f32) + C(16x16 f32), exact fp32 FMA path
  return __builtin_amdgcn_wmma_f32_16x16x4_f32(
      /*neg_a=*/false, a, /*neg_b=*/false, b,
      /*c_mod=*/(short)0, c, /*reuse_a=*/false, /*reuse_b=*/false);
}

// ---------------------------------------------------------------- utilities
__global__ void zero_kernel(float* __restrict__ p, int n) {
  int i = blockIdx.x * blockDim.x + threadIdx.x;
  int stride = gridDim.x * blockDim.x;
  for (; i < n; i += stride) p[i] = 0.0f;
}

// ------------------------------------------------------------ h = x@W_in + b
__global__ void input_gemm_kernel(const float* __restrict__ x,
                                  const float* __restrict__ W,
                                  const float* __restrict__ b,
                                  float* __restrict__ h) {
  __shared__ float xs[IN_DIM];
  const int nid = blockIdx.x;
  const int t = threadIdx.x;
  if (t < IN_DIM) xs[t] = x[(size_t)nid * IN_DIM + t];
  __syncthreads();
#pragma unroll
  for (int jj = 0; jj < 2; ++jj) {
    const int j = t + jj * 256;
    float s = b[j];
#pragma unroll
    for (int k = 0; k < IN_DIM; ++k) s = fmaf(xs[k], W[k * H_DIM + j], s);
    h[(size_t)nid * H_DIM + j] = s;
  }
}

// ----------------------------------------------- agg[dst] += h[src] per edge
__global__ void scatter_kernel(const float* __restrict__ h,
                               const int* __restrict__ ei,
                               float* __restrict__ agg) {
  const int wave = (blockIdx.x * blockDim.x + threadIdx.x) >> 5;
  const int lane = threadIdx.x & 31;
  if (wave >= E_EDGES) return;
  const int s = ei[wave];
  const int d = ei[E_EDGES + wave];
  const float* __restrict__ hs = h + (size_t)s * H_DIM;
  float* __restrict__ ad = agg + (size_t)d * H_DIM;
#pragma unroll
  for (int t = 0; t < H_DIM / 32; ++t)
    atomicAdd(&ad[t * 32 + lane], hs[t * 32 + lane]);
}

// ---- fused: h = relu(LN(h@Wr + agg@Wn + bias)) ; one block = 16 rows x 512
__global__ void __launch_bounds__(1024)
conv_layer_kernel(float* __restrict__ h, const float* __restrict__ agg,
                  const float* __restrict__ Wr, const float* __restrict__ Wn,
                  const float* __restrict__ bias,
                  const float* __restrict__ gamma,
                  const float* __restrict__ beta) {
  __shared__ float tile[16 * 516];  // 16 rows, stride 516 to avoid conflicts
  const int rowbase = blockIdx.x * 16;
  const int wave = threadIdx.x >> 5;   // 0..31 -> column tile
  const int lane = threadIdx.x & 31;
  const int col0 = wave * 16;
  const int m = lane & 15;             // A row within tile
  const int koff = (lane >> 4) * 2;    // A/B K sub-offset per half-wave
  const int n = lane & 15;             // B/C column within tile

  const float* __restrict__ hrow = h + (size_t)(rowbase + m) * H_DIM;
  const float* __restrict__ grow = agg + (size_t)(rowbase + m) * H_DIM;

  v8f acc = {0.f, 0.f, 0.f, 0.f, 0.f, 0.f, 0.f, 0.f};
#pragma unroll 4
  for (int k0 = 0; k0 < H_DIM; k0 += 4) {
    const v2f ah = *(const v2f*)(hrow + k0 + koff);
    const v2f ag = *(const v2f*)(grow + k0 + koff);
    v2f br, bn;
    br.x = Wr[(size_t)(k0 + koff) * H_DIM + col0 + n];
    br.y = Wr[(size_t)(k0 + koff + 1) * H_DIM + col0 + n];
    bn.x = Wn[(size_t)(k0 + koff) * H_DIM + col0 + n];
    bn.y = Wn[(size_t)(k0 + koff + 1) * H_DIM + col0 + n];
    acc = wmma_f32_k4(ah, br, acc);
    acc = wmma_f32_k4(ag, bn, acc);
  }

  // Stage (acc + bias) into LDS: VGPR i -> row i (lanes 0-15) / row i+8
  const float bval = bias[col0 + n];
#pragma unroll
  for (int i = 0; i < 8; ++i) {
    const int row = i + (lane >> 4) * 8;
    tile[row * 516 + col0 + n] = acc[i] + bval;
  }
  __syncthreads();

  // LayerNorm + ReLU: wave w (<16) owns row w; 16 elems/lane, shfl reduce
  if (wave < 16) {
    const int r = wave;
    float v[16];
    float s = 0.f, sq = 0.f;
#pragma unroll
    for (int t = 0; t < 16; ++t) {
      v[t] = tile[r * 516 + t * 32 + lane];
      s += v[t];
      sq += v[t] * v[t];
    }
#pragma unroll
    for (int off = 16; off > 0; off >>= 1) {
      s  += __shfl_xor(s, off, 32);
      sq += __shfl_xor(sq, off, 32);
    }
    const float mu  = s * (1.0f / H_DIM);
    const float var = sq * (1.0f / H_DIM) - mu * mu;
    const float inv = rsqrtf(var + LN_EPS);
    float* __restrict__ orow = h + (size_t)(rowbase + r) * H_DIM;
#pragma unroll
    for (int t = 0; t < 16; ++t) {
      const int c = t * 32 + lane;
      const float y = (v[t] - mu) * inv * gamma[c] + beta[c];
      orow[c] = fmaxf(y, 0.0f);
    }
  }
}

// --------------------------------------------- pooled[batch[n]] += h[n]
__global__ void pool_scatter_kernel(const float* __restrict__ h,
                                    const int* __restrict__ batch,
                                    float* __restrict__ pooled,
                                    float* __restrict__ counts) {
  const int wave = (blockIdx.x * blockDim.x + threadIdx.x) >> 5;
  const int lane = threadIdx.x & 31;
  if (wave >= N_NODES) return;
  const int g = batch[wave];
  const float* __restrict__ hr = h + (size_t)wave * H_DIM;
  float* __restrict__ pg = pooled + (size_t)g * H_DIM;
#pragma unroll
  for (int t = 0; t < H_DIM / 32; ++t)
    atomicAdd(&pg[t * 32 + lane], hr[t * 32 + lane]);
  if (lane == 0) atomicAdd(&counts[g], 1.0f);
}

__global__ void pool_div_kernel(float* __restrict__ pooled,
                                const float* __restrict__ counts) {
  const int i = blockIdx.x * blockDim.x + threadIdx.x;
  if (i >= G_GRAPHS * H_DIM) return;
  pooled[i] /= fmaxf(counts[i / H_DIM], 1.0f);
}

// ----------------------------------- out = pooled @ W_out + b_out (256x512)
__global__ void __launch_bounds__(1024)
out_gemm_kernel(const float* __restrict__ pooled, const float* __restrict__ W,
                const float* __restrict__ b, float* __restrict__ out) {
  const int rowbase = blockIdx.x * 16;
  const int wave = threadIdx.x >> 5;
  const int lane = threadIdx.x & 31;
  const int col0 = wave * 16;
  const int m = lane & 15;
  const int koff = (lane >> 4) * 2;
  const int n = lane & 15;

  const float* __restrict__ prow = pooled + (size_t)(rowbase + m) * H_DIM;
  v8f acc = {0.f, 0.f, 0.f, 0.f, 0.f, 0.f, 0.f, 0.f};
#pragma unroll 4
  for (int k0 = 0; k0 < H_DIM; k0 += 4) {
    const v2f a = *(const v2f*)(prow + k0 + koff);
    v2f bb;
    bb.x = W[(size_t)(k0 + koff) * H_DIM + col0 + n];
    bb.y = W[(size_t)(k0 + koff + 1) * H_DIM + col0 + n];
    acc = wmma_f32_k4(a, bb, acc);
  }
  const float bv = b[col0 + n];
#pragma unroll
  for (int i = 0; i < 8; ++i) {
    const int row = rowbase + i + (lane >> 4) * 8;
    out[(size_t)row * H_DIM + col0 + n] = acc[i] + bv;
  }
}

extern "C" void kernel_launch(void* const* d_in, const int* in_sizes, int n_in,
                              void* d_out, int out_size, void* d_ws, size_t ws_size,
                              hipStream_t stream) {
  const float* x        = (const float*)d_in[0];
  const int*   ei       = (const int*)  d_in[1];   // [2, E] (src row, dst row)
  const int*   batch    = (const int*)  d_in[2];   // [N]
  const float* W_in     = (const float*)d_in[3];
  const float* b_in     = (const float*)d_in[4];
  const float* W_root   = (const float*)d_in[5];   // [L, H, H]
  const float* W_neigh  = (const float*)d_in[6];   // [L, H, H]
  const float* b_conv   = (const float*)d_in[7];   // [L, H]
  const float* ln_gamma = (const float*)d_in[8];   // [L, H]
  const float* ln_beta  = (const float*)d_in[9];   // [L, H]
  const float* W_out    = (const float*)d_in[10];
  const float* b_out    = (const float*)d_in[11];
  float* out = (float*)d_out;

  float* h      = (float*)d_ws;                    // N*H
  float* agg    = h + (size_t)N_NODES * H_DIM;     // N*H
  float* pooled = agg + (size_t)N_NODES * H_DIM;   // G*H
  float* counts = pooled + (size_t)G_GRAPHS * H_DIM; // G

  // h = x @ W_in + b_in
  input_gemm_kernel<<<N_NODES, 256, 0, stream>>>(x, W_in, b_in, h);

  for (int l = 0; l < L_LAYERS; ++l) {
    zero_kernel<<<2048, 256, 0, stream>>>(agg, N_NODES * H_DIM);
    scatter_kernel<<<E_EDGES / 8, 256, 0, stream>>>(h, ei, agg);
    conv_layer_kernel<<<N_NODES / 16, 1024, 0, stream>>>(
        h, agg,
        W_root  + (size_t)l * H_DIM * H_DIM,
        W_neigh + (size_t)l * H_DIM * H_DIM,
        b_conv  + (size_t)l * H_DIM,
        ln_gamma + (size_t)l * H_DIM,
        ln_beta  + (size_t)l * H_DIM);
  }

  zero_kernel<<<512, 256, 0, stream>>>(pooled, G_GRAPHS * H_DIM + G_GRAPHS);
  pool_scatter_kernel<<<N_NODES / 8, 256, 0, stream>>>(h, batch, pooled, counts);
  pool_div_kernel<<<(G_GRAPHS * H_DIM + 255) / 256, 256, 0, stream>>>(pooled, counts);
  out_gemm_kernel<<<G_GRAPHS / 16, 1024, 0, stream>>>(pooled, W_out, b_out, out);
}